// MeMCeNNLayer_80839874445472
// MI455X (gfx1250) — compile-verified
//
#include <hip/hip_runtime.h>
#include <stdint.h>

// ---------------- problem constants ----------------
#define B_   32
#define H_   1024
#define W_   1024
#define WIN_ 1025          // input row pitch (before slicing off column 0)

#define TH_   64
#define TW_   64
#define HALO_ 3
#define EH_   70           // extended tile rows
#define EW_   70           // extended tile cols
#define NT_   256          // 8 wave32 per workgroup

// dense stage buffer (TDM target), 70x70 row-major + small tail guard
#define STG_  (EH_*EW_)              // 4900
#define STGA_ (STG_ + 4)
#define SPT_  ((STG_ + NT_-1)/NT_)   // 20 (per-lane fallback staging)

// bordered working arrays: 72 rows x 76-dword pitch, data rows 1..70, cols 4..73
#define SP_   76
#define ROWS_ 72
#define PAD_  (ROWS_*SP_)            // 5472 dwords = 21.9 KB
#define C0_   4                      // first data column (16B aligned runs)
#define RPW_  18                     // runs-of-4 per data row (covers cols 4..75)
#define NRUN_ (EH_*RPW_)             // 1260
#define RPT_  ((NRUN_ + NT_-1)/NT_)  // 5 runs per thread

#define DT_     0.3f
#define ALPHA_  0.01f
#define Zc_     (-1.9875f)

// polynomial constants
#define cA0 (-0.9602f)
#define cA1 (-0.628f)
#define cA2 (-2.1831f)
#define cA3 (1.5359f)
#define cB1 (1.6686f)
#define cB2 (1.2363f)
#define cB3 (-1.1509f)
#define cC1 (0.5227f)
#define cC2 (-1.8543f)
#define cC3 (-0.045f)
#define cD0 (-1.204f)
#define cD1 (-1.2903f)
#define cD2 (-1.5158f)
#define cD3 (-0.221f)
#define cE1 (0.6489f)
#define cE2 (0.8607f)
#define cE3 (-0.8444f)
#define cF1 (0.2612f)
#define cF2 (-0.4397f)
#define cF3 (1.0842f)

// constexpr so taps fold to inline literals (no constant-memory loads)
constexpr float WAc[3][3] = {
    {-0.1176f,  0.0788f, -0.1068f},
    {-0.7379f,  1.1183f, -0.8916f},
    { 0.1008f, -0.1443f,  0.1355f}};
constexpr float WBc[3][3] = {
    { 0.0557f,  0.1734f,  0.2607f},
    {-1.5312f,  0.6234f, -1.2246f},
    {-0.2078f, -0.2330f, -0.0756f}};

// ---------------- CDNA5 TDM availability ----------------
#if defined(__gfx1250__) && __has_builtin(__builtin_amdgcn_tensor_load_to_lds)
#define USE_TDM 1
#else
#define USE_TDM 0
#endif

#if defined(__gfx1250__) && !USE_TDM
#warning "CDNA5: __builtin_amdgcn_tensor_load_to_lds NOT available; per-lane staging in use"
#endif

#if USE_TDM
typedef unsigned int tdm_v4u __attribute__((ext_vector_type(4)));
typedef int          tdm_v4i __attribute__((ext_vector_type(4)));
typedef int          tdm_v8i __attribute__((ext_vector_type(8)));
#endif

__global__ __launch_bounds__(NT_) void memcenn_fused(
    const float* __restrict__ x,
    const float* __restrict__ mem_init,
    float* __restrict__ outp,
    float* __restrict__ statesp)
{
    // s_pool: first acts as dense TDM stage, then reused as s_ic (stage is dead)
    __shared__ float s_pool[PAD_];
    __shared__ float s_a[PAD_];   // evolving x (bordered layout)
    __shared__ float s_o[PAD_];   // leaky(x) conv input (bordered layout)
    float* const s_stage = s_pool;
    float* const s_ic    = s_pool;

    const int tid = threadIdx.x;
    const int b   = blockIdx.z;
    const int th0 = blockIdx.y * TH_;
    const int tw0 = blockIdx.x * TW_;
    const int gh0 = th0 - HALO_;
    const int gw0 = tw0 - HALO_;
    const float* __restrict__ xb = x + (size_t)b * H_ * WIN_;

    const bool interior = (gh0 >= 0) & (gw0 >= 0) &
                          (gh0 + EH_ <= H_) & (gw0 + EW_ <= W_);

    // ---------- stage x tile (dense 70x70) into LDS ----------
#if USE_TDM
    if (interior) {
        if ((tid >> 5) == 0) {  // one wave issues the TDM DMA (EXEC-independent)
            const uint64_t ga   = (uint64_t)(uintptr_t)(xb + (size_t)gh0 * WIN_ + (gw0 + 1));
            const uint32_t ldsa = (uint32_t)(uintptr_t)(&s_stage[0]);
            tdm_v4u g0;                               // D# group 0
            g0.x = 1u;                                // count=1 valid descriptor
            g0.y = ldsa;                              // lds_addr
            g0.z = (uint32_t)(ga & 0xffffffffull);
            g0.w = (uint32_t)((ga >> 32) & 0x1ffffffull) | (2u << 30);  // type=2
            tdm_v8i g1;                               // D# group 1
            g1[0] = 0x00020000;                       // data_size=4B
            g1[1] = (int)((unsigned)WIN_ << 16);      // tensor_dim0 = 1025 (lo16)
            g1[2] = (int)((unsigned)H_  << 16);       // dim0 hi=0 | tensor_dim1=1024 lo16
            g1[3] = (int)((unsigned)EW_ << 16);       // dim1 hi=0 | tile_dim0=70
            g1[4] = EH_;                              // tile_dim1=70, tile_dim2=0
            g1[5] = WIN_;                             // tensor_dim0_stride = 1025
            g1[6] = 0;
            g1[7] = 0;
            tdm_v4i g2 = {0, 0, 0, 0};
            tdm_v4i g3 = {0, 0, 0, 0};
#if __clang_major__ >= 23
            tdm_v8i g4 = {0, 0, 0, 0, 0, 0, 0, 0};
            __builtin_amdgcn_tensor_load_to_lds(g0, g1, g2, g3, g4, 0);
#else
            __builtin_amdgcn_tensor_load_to_lds(g0, g1, g2, g3, 0);
#endif
#if __has_builtin(__builtin_amdgcn_s_wait_tensorcnt)
            __builtin_amdgcn_s_wait_tensorcnt(0);
#else
            asm volatile("s_wait_tensorcnt 0x0" ::: "memory");
#endif
        }
    } else
#endif
    {
        // boundary tiles (or no-TDM fallback): guarded per-lane loads, zero pad
#pragma unroll
        for (int j = 0; j < SPT_; ++j) {
            int i = tid + j * NT_;
            if (i < STG_) {
                int er = i / EW_, ec = i - er * EW_;
                int gh = gh0 + er, gw = gw0 + ec;
                float v = 0.f;
                if (gh >= 0 && gh < H_ && gw >= 0 && gw < W_)
                    v = xb[(size_t)gh * WIN_ + (gw + 1)];
                s_stage[i] = v;
            }
        }
    }
    __syncthreads();

    // ---------- relayout stage -> bordered s_a, init mem registers ----------
    float memv[RPT_][4];
#pragma unroll
    for (int j = 0; j < RPT_; ++j) {
        int rid = tid + j * NT_;
#pragma unroll
        for (int t = 0; t < 4; ++t) memv[j][t] = 0.f;
        if (rid < NRUN_) {
            int er  = rid / RPW_;                    // extended row 0..69
            int cb  = C0_ + (rid - er * RPW_) * 4;   // LDS col base, 16B aligned
            int row = er + 1;
            int base = row * SP_ + cb;
#pragma unroll
            for (int t = 0; t < 4; ++t) {
                int exc = cb - C0_ + t;              // extended col 0..71 (70,71 = pad)
                s_a[base + t] = s_stage[er * EW_ + exc];  // tail reads guarded by STGA_
                int gw = gw0 + exc;
                gw = gw < 0 ? 0 : (gw > W_ - 1 ? W_ - 1 : gw);
                memv[j][t] = mem_init[gw];
            }
        }
    }
    __syncthreads();   // stage fully consumed; s_pool may now be reused as s_ic

    // ---------- input_coupling = conv(x0, Wb) + Z  (bordered, affine taps) ----------
#pragma unroll
    for (int j = 0; j < RPT_; ++j) {
        int rid = tid + j * NT_;
        if (rid < NRUN_) {
            int er  = rid / RPW_;
            int cb  = C0_ + (rid - er * RPW_) * 4;
            int row = er + 1;
            float w0[6], w1[6], w2[6];
#pragma unroll
            for (int u = 0; u < 6; ++u) {
                w0[u] = s_a[(row - 1) * SP_ + cb - 1 + u];
                w1[u] = s_a[ row      * SP_ + cb - 1 + u];
                w2[u] = s_a[(row + 1) * SP_ + cb - 1 + u];
            }
            int base = row * SP_ + cb;
#pragma unroll
            for (int t = 0; t < 4; ++t) {
                float acc = Zc_;
#pragma unroll
                for (int dc = 0; dc < 3; ++dc)
                    acc += WBc[0][dc] * w0[t + dc] + WBc[1][dc] * w1[t + dc] +
                           WBc[2][dc] * w2[t + dc];
                s_ic[base + t] = acc;
            }
        }
    }
    __syncthreads();

    const size_t imgBase = (size_t)b * H_ * W_;

    for (int k = 0; k < 3; ++k) {
        // ---- phase 1: pointwise membrane dynamics ----
#pragma unroll
        for (int j = 0; j < RPT_; ++j) {
            int rid = tid + j * NT_;
            if (rid < NRUN_) {
                int er  = rid / RPW_;
                int cb  = C0_ + (rid - er * RPW_) * 4;
                int row = er + 1;
                int base = row * SP_ + cb;
                int gh = gh0 + er;
                bool rowin  = (gh >= 0) & (gh < H_);
                bool rowctr = (er >= HALO_) & (er < EH_ - HALO_);
#pragma unroll
                for (int t = 0; t < 4; ++t) {
                    int idx = base + t;
                    float Vm = s_a[idx];
                    float m  = memv[j][t];
                    float P1 = cA0 + Vm * (cA1 + Vm * (cA2 + Vm * cA3));
                    float P2 = m * (cB1 + m * (cB2 + m * cB3));
                    float P3 = Vm * m * (cC1 + cC2 * m + cC3 * Vm);
                    m = m + DT_ * (P1 + P2 + P3);
                    memv[j][t] = m;
                    float V1 = cD0 + Vm * (cD1 + Vm * (cD2 + Vm * cD3));
                    float V2 = m * (cE1 + m * (cE2 + m * cE3));
                    float V3 = Vm * m * (cF1 + cF2 * m + cF3 * Vm);
                    float Imem = Vm * (V1 + V2 + V3);

                    float y  = fminf(Vm, 1.0f + ALPHA_ * (Vm - 1.0f));
                    float ov = fmaxf(y, -1.0f + ALPHA_ * (y + 1.0f));
                    int exc = cb - C0_ + t;
                    int gw  = gw0 + exc;
                    bool inimg = rowin & (gw >= 0) & (gw < W_);
                    if (!inimg) ov = 0.f;            // SAME zero-padding semantics
                    s_o[idx] = ov;
                    // R = x + DT*(ic - Imem - x); conv term added in phase 2
                    s_a[idx] = Vm + DT_ * (s_ic[idx] - Imem - Vm);

                    bool center = rowctr & (exc >= HALO_) & (exc < EW_ - HALO_);
                    if (k == 2 && center)            // out = leaky(x2)
                        outp[imgBase + (size_t)gh * W_ + gw] = ov;
                }
            }
        }
        __syncthreads();

        // ---- phase 2: coupling conv + x update, emit States ----
#pragma unroll
        for (int j = 0; j < RPT_; ++j) {
            int rid = tid + j * NT_;
            if (rid < NRUN_) {
                int er  = rid / RPW_;
                int cb  = C0_ + (rid - er * RPW_) * 4;
                int row = er + 1;
                float w0[6], w1[6], w2[6];
#pragma unroll
                for (int u = 0; u < 6; ++u) {
                    w0[u] = s_o[(row - 1) * SP_ + cb - 1 + u];
                    w1[u] = s_o[ row      * SP_ + cb - 1 + u];
                    w2[u] = s_o[(row + 1) * SP_ + cb - 1 + u];
                }
                int base = row * SP_ + cb;
                int gh = gh0 + er;
                bool rowctr = (er >= HALO_) & (er < EH_ - HALO_);
#pragma unroll
                for (int t = 0; t < 4; ++t) {
                    float acc = 0.f;
#pragma unroll
                    for (int dc = 0; dc < 3; ++dc)
                        acc += WAc[0][dc] * w0[t + dc] + WAc[1][dc] * w1[t + dc] +
                               WAc[2][dc] * w2[t + dc];
                    float xn = s_a[base + t] + DT_ * acc;
                    s_a[base + t] = xn;
                    int exc = cb - C0_ + t;
                    bool center = rowctr & (exc >= HALO_) & (exc < EW_ - HALO_);
                    if (center) {
                        int gw = gw0 + exc;
                        statesp[(imgBase + (size_t)gh * W_ + gw) * 3 + k] = xn;
                    }
                }
            }
        }
        __syncthreads();
    }
}

extern "C" void kernel_launch(void* const* d_in, const int* in_sizes, int n_in,
                              void* d_out, int out_size, void* d_ws, size_t ws_size,
                              hipStream_t stream) {
    const float* x        = (const float*)d_in[0];   // (32,1,1024,1025)
    const float* mem_init = (const float*)d_in[3];   // (1,1,1,1024)
    float* outp    = (float*)d_out;                  // (32,1,1024,1024)
    float* statesp = outp + (size_t)B_ * H_ * W_;    // (32,1,1024,1024,3)
    dim3 grid(W_ / TW_, H_ / TH_, B_);               // (16,16,32)
    memcenn_fused<<<grid, NT_, 0, stream>>>(x, mem_init, outp, statesp);
}